// NWJ_14422500180348
// MI455X (gfx1250) — compile-verified
//
#include <hip/hip_runtime.h>
#include <math.h>

typedef __attribute__((ext_vector_type(2))) float v2f;
typedef __attribute__((ext_vector_type(8))) float v8f;

#define N_SAMP 1024
#define XD     512
#define HID    256

// ---------------------------------------------------------------------------
// Kernel 1: fp32 WMMA GEMM.  out(1024x256) = A(1024x512) @ W(512x256)
// One wave32 per 16x16 output tile; K accumulated in steps of 4 via
// V_WMMA_F32_16X16X4_F32.  t = global tile id; which selects (X,Wx)->px or
// (Y,Wy)->py.  2048 tiles total = 256 blocks * 8 waves.
//
// Fragment layouts (ISA 7.12.2):
//   A 16x4 f32 : lane = h*16+m (h=lane>>4), a[v] = A[m, k0 + 2*h + v]
//   B 4x16 f32 : lane = h*16+n,             b[v] = W[k0 + 2*h + v, n]
//   C 16x16    : vgpr r, lane half h -> row r + 8*h, col = lane&15
// ---------------------------------------------------------------------------
__global__ __launch_bounds__(256) void nwj_gemm_wmma(
    const float* __restrict__ X, const float* __restrict__ Y,
    const float* __restrict__ W1, float* __restrict__ px, float* __restrict__ py)
{
    const int lane = threadIdx.x & 31;
    const int wave = threadIdx.x >> 5;
    const int t    = blockIdx.x * 8 + wave;   // 0..2047
    const int which = t >> 10;                // 0 -> px, 1 -> py
    const int rem   = t & 1023;
    const int tm    = rem >> 4;               // row tile 0..63
    const int tn    = rem & 15;               // col tile 0..15

    const float* A = which ? Y : X;
    const float* W = W1 + which * (XD * HID); // Wx = W1[:512], Wy = W1[512:]
    float* out     = which ? py : px;

    const int m   = lane & 15;
    const int h   = lane >> 4;
    const int col = tn * 16 + m;
    const float* Arow = A + (size_t)(tm * 16 + m) * XD;

    v8f c = {0.f, 0.f, 0.f, 0.f, 0.f, 0.f, 0.f, 0.f};
#pragma unroll 4
    for (int k = 0; k < XD; k += 4) {
        v2f a, b;
        a.x = Arow[k + 2 * h];
        a.y = Arow[k + 2 * h + 1];
        b.x = W[(k + 2 * h) * HID + col];
        b.y = W[(k + 2 * h + 1) * HID + col];
        c = __builtin_amdgcn_wmma_f32_16x16x4_f32(
                /*neg_a=*/false, a, /*neg_b=*/false, b,
                /*c_mod=*/(short)0, c, /*reuse_a=*/false, /*reuse_b=*/false);
    }

    float* Cb = out + (size_t)(tm * 16) * HID + col;
#pragma unroll
    for (int r = 0; r < 8; ++r)
        Cb[(size_t)(r + 8 * h) * HID] = c[r];
}

// ---------------------------------------------------------------------------
// Kernel 2: pairwise ReLU-dot + per-row logsumexp.
// Block bi handles rows i = 8*bi .. 8*bi+7.  py rows (+b1) and w2 staged in
// LDS; px streamed as float4 (L2-resident).  t1 rows kept in LDS (32 KB),
// then each of the 8 waves reduces one row (max + sum-exp) with shuffles.
// T0[i] = t1[i][i] (the j==i diagonal equals T0 before the -1 shift).
// ---------------------------------------------------------------------------
__global__ __launch_bounds__(256) void nwj_pairwise_lse(
    const float* __restrict__ px, const float* __restrict__ py,
    const float* __restrict__ b1, const float* __restrict__ w2,
    const float* __restrict__ b2, float* __restrict__ T0, float* __restrict__ E)
{
    __shared__ __align__(16) float pyb[8][HID];
    __shared__ __align__(16) float w2s[HID];
    __shared__ __align__(16) float t1[8][N_SAMP];

    const int tid = threadIdx.x;
    const int bi  = blockIdx.x;
    const float b2v = b2[0];

    for (int idx = tid; idx < 8 * HID; idx += 256) {
        int ii = idx >> 8, k = idx & (HID - 1);
        pyb[ii][k] = py[(size_t)(bi * 8 + ii) * HID + k] + b1[k];
    }
    w2s[tid] = w2[tid];
    __syncthreads();

    for (int jj = 0; jj < 4; ++jj) {
        const int j = jj * 256 + tid;
        const float4* pxj = (const float4*)(px + (size_t)j * HID);
        float acc[8];
#pragma unroll
        for (int ii = 0; ii < 8; ++ii) acc[ii] = 0.f;

        for (int k4 = 0; k4 < HID / 4; ++k4) {
            const float4 p = pxj[k4];
            const float4 w = ((const float4*)w2s)[k4];
#pragma unroll
            for (int ii = 0; ii < 8; ++ii) {
                const float4 q = *(const float4*)&pyb[ii][k4 * 4];
                acc[ii] = fmaf(fmaxf(p.x + q.x, 0.f), w.x, acc[ii]);
                acc[ii] = fmaf(fmaxf(p.y + q.y, 0.f), w.y, acc[ii]);
                acc[ii] = fmaf(fmaxf(p.z + q.z, 0.f), w.z, acc[ii]);
                acc[ii] = fmaf(fmaxf(p.w + q.w, 0.f), w.w, acc[ii]);
            }
        }
#pragma unroll
        for (int ii = 0; ii < 8; ++ii) t1[ii][j] = acc[ii] + b2v;
    }
    __syncthreads();

    const int wv = tid >> 5, lane = tid & 31;   // wave wv reduces row wv
    float m = -INFINITY;
    for (int j = lane; j < N_SAMP; j += 32) m = fmaxf(m, t1[wv][j]);
#pragma unroll
    for (int off = 16; off > 0; off >>= 1) m = fmaxf(m, __shfl_xor(m, off, 32));
    float s = 0.f;
    for (int j = lane; j < N_SAMP; j += 32) s += __expf(t1[wv][j] - m);
#pragma unroll
    for (int off = 16; off > 0; off >>= 1) s += __shfl_xor(s, off, 32);

    if (lane == 0) {
        const int gi = bi * 8 + wv;
        // exp(lse_i) with T1 = t1raw - 1 and the 1/n inside logsumexp:
        E[gi]  = __expf(m - 1.0f) * s * (1.0f / (float)N_SAMP);
        T0[gi] = t1[wv][gi];
    }
}

// ---------------------------------------------------------------------------
// Kernel 3: loss = mean(E) - mean(T0)  (= -(T0.mean() - exp(lse).mean()))
// ---------------------------------------------------------------------------
__global__ __launch_bounds__(256) void nwj_final_reduce(
    const float* __restrict__ T0, const float* __restrict__ E,
    float* __restrict__ out)
{
    __shared__ float red[256];
    const int tid = threadIdx.x;
    float s = 0.f;
    for (int i = tid; i < N_SAMP; i += 256) s += (E[i] - T0[i]);
    red[tid] = s;
    __syncthreads();
    for (int off = 128; off > 0; off >>= 1) {
        if (tid < off) red[tid] += red[tid + off];
        __syncthreads();
    }
    if (tid == 0) out[0] = red[0] * (1.0f / (float)N_SAMP);
}

extern "C" void kernel_launch(void* const* d_in, const int* in_sizes, int n_in,
                              void* d_out, int out_size, void* d_ws, size_t ws_size,
                              hipStream_t stream)
{
    const float* x  = (const float*)d_in[0];  // 1024x512
    const float* y  = (const float*)d_in[1];  // 1024x512
    const float* W1 = (const float*)d_in[2];  // 1024x256
    const float* b1 = (const float*)d_in[3];  // 256
    const float* W2 = (const float*)d_in[4];  // 256x1
    const float* b2 = (const float*)d_in[5];  // 1

    float* px = (float*)d_ws;                 // 1024*256
    float* py = px + N_SAMP * HID;            // 1024*256
    float* T0 = py + N_SAMP * HID;            // 1024
    float* E  = T0 + N_SAMP;                  // 1024

    nwj_gemm_wmma<<<256, 256, 0, stream>>>(x, y, W1, px, py);
    nwj_pairwise_lse<<<128, 256, 0, stream>>>(px, py, b1, W2, b2, T0, E);
    nwj_final_reduce<<<1, 256, 0, stream>>>(T0, E, (float*)d_out);
}